// VectorQuantizer_17145509446289
// MI455X (gfx1250) — compile-verified
//
#include <hip/hip_runtime.h>
#include <hip/hip_bf16.h>

typedef __attribute__((ext_vector_type(2))) float v2f;
typedef __attribute__((ext_vector_type(8))) float v8f;

#define H_IMG 2048
#define W_IMG 2048
#define C_IMG 3
#define D_FEAT 192           // 8*8*3
#define N_CODE 4096
#define L_ROWS 65536         // (2048/8)^2
#define M_TILE 64            // rows per workgroup
#define N_STEP 64            // codebook columns per iteration
#define N_ITERS (N_CODE / N_STEP)   // 64
#define STRIDE 196           // LDS row stride in floats: 196%64==4 -> conflict-free, 16B aligned
#define THREADS 512          // 16 waves (wave32)

// ---- codebook squared norms: c2[n] = sum_d cb[n,d]^2 ----
__global__ void vq_c2_kernel(const float* __restrict__ cb, float* __restrict__ c2) {
    int n = blockIdx.x * blockDim.x + threadIdx.x;
    if (n < N_CODE) {
        const float* row = cb + n * D_FEAT;
        float s = 0.f;
#pragma unroll 4
        for (int d = 0; d < D_FEAT; d += 4) {
            float4 v = *(const float4*)(row + d);
            s = fmaf(v.x, v.x, s);
            s = fmaf(v.y, v.y, s);
            s = fmaf(v.z, v.z, s);
            s = fmaf(v.w, v.w, s);
        }
        c2[n] = s;
    }
}

// Map (feature row l, feature dim d) -> flat image index, replicating the
// reference's DOUBLE blockify (b2 = blockify(blockify(image))) element order.
__device__ __forceinline__ int vq_src_index(int l, int d) {
    int c2 = d >> 6;             // d / 64
    int rem = d & 63;
    int i2 = rem >> 3;
    int j2 = rem & 7;
    int l1 = (l >> 3) * 8 + i2;  // b1 row feeding this element
    int hb = l1 >> 8;            // / 256
    int wb = l1 & 255;
    int f = (l & 7) * 24 + j2 * 3 + c2;  // = p1*3 + c1 = c*64 + i*8 + j
    int c = f >> 6;
    int i = (f >> 3) & 7;
    int j = f & 7;
    return ((hb * 8 + i) * W_IMG + (wb * 8 + j)) * C_IMG + c;
}

// Issue one iteration's codebook tile as async global->LDS b128 copies.
// 64 rows * 48 b128-chunks = 3072 lane-slots = exactly 6 per thread
// => every wave issues exactly 6 async instructions (uniform ASYNCcnt).
__device__ __forceinline__ void vq_issue_btile(const float* __restrict__ cb,
                                               int nbase, float* ldsbase, int tid) {
#pragma unroll
    for (int q = 0; q < 6; ++q) {
        int s = tid + q * THREADS;           // [0, 3072)
        int n = s / 48;                      // row within tile  [0,64)
        int c = s - n * 48;                  // b128 chunk (4 floats) [0,48)
        unsigned loff = (unsigned)(size_t)(ldsbase + n * STRIDE + 4 * c);
        unsigned long long g =
            (unsigned long long)(size_t)(cb + (size_t)(nbase + n) * D_FEAT + 4 * c);
        asm volatile("global_load_async_to_lds_b128 %0, %1, off"
                     :: "v"(loff), "v"(g) : "memory");
    }
}

// ---- fused GEMM (X * CB^T via fp32 WMMA) + running argmin + gather-out ----
__global__ __launch_bounds__(THREADS, 1) void vq_main_kernel(
        const float* __restrict__ image,
        const float* __restrict__ cb,
        const float* __restrict__ c2arr,
        float* __restrict__ out) {
    __shared__ __align__(16) float xlds[M_TILE * STRIDE];          // ~49 KB  (X tile)
    __shared__ __align__(16) float blds[2][N_STEP * STRIDE];       // ~98 KB  (double-buffered B)
    __shared__ float redS[16 * 8 * 32];                            // 16 KB
    __shared__ int   redI[16 * 8 * 32];                            // 16 KB
    __shared__ int   rowIdx[M_TILE];

    const int tid = threadIdx.x;
    const int lbase = blockIdx.x * M_TILE;

    // Prefetch iteration 0's codebook tile while we gather X.
    vq_issue_btile(cb, 0, &blds[0][0], tid);

    // Gather permuted X tile into LDS (each image element read exactly once device-wide).
    for (int e = tid; e < M_TILE * D_FEAT; e += THREADS) {
        int lr = e / D_FEAT;
        int d  = e - lr * D_FEAT;
        xlds[lr * STRIDE + d] = image[vq_src_index(lbase + lr, d)];
    }
    __syncthreads();

    const int lane = tid & 31;
    const int wv   = tid >> 5;       // wave id 0..15
    const int rt   = wv >> 2;        // row-tile 0..3 (16 rows each)
    const int ct   = wv & 3;         // col-tile 0..3 (16 codes each)
    const int m    = lane & 15;
    const int hi   = lane >> 4;      // K-pair select within WMMA A/B layout

    // Preload all 48 A operand pairs for this wave's 16 rows into registers.
    // Lane m holds row (rt*16+m); VGPR0/1 = K {0,1} (lanes 0-15) or {2,3} (lanes 16-31).
    const float* arow = &xlds[(rt * 16 + m) * STRIDE + 2 * hi];
    v2f areg[48];
#pragma unroll
    for (int t = 0; t < 48; ++t) areg[t] = *(const v2f*)(arow + 4 * t);

    float best[8];
    int   bidx[8];
#pragma unroll
    for (int r = 0; r < 8; ++r) { best[r] = 3.4e38f; bidx[r] = 0; }

    for (int it = 0; it < N_ITERS; ++it) {
        // Prefetch next tile into the other buffer, then wait for current tile.
        if (it + 1 < N_ITERS) {
            vq_issue_btile(cb, (it + 1) * N_STEP, &blds[(it + 1) & 1][0], tid);
            asm volatile("s_wait_asynccnt 0x6" ::: "memory");   // newest 6 outstanding => batch(it) done
        } else {
            asm volatile("s_wait_asynccnt 0x0" ::: "memory");
        }
        __syncthreads();   // all waves' async writes for batch(it) visible

        const int nlane = it * N_STEP + ct * 16 + m;   // this lane's codebook column
        const float* brow = &blds[it & 1][(ct * 16 + m) * STRIDE + 2 * hi];

        // Two independent 24-deep accumulation chains for ILP.
        v8f acc0 = {};
        v8f acc1 = {};
#pragma unroll
        for (int t = 0; t < 24; ++t) {
            v2f b = *(const v2f*)(brow + 4 * t);
            acc0 = __builtin_amdgcn_wmma_f32_16x16x4_f32(false, areg[t], false, b,
                                                         (short)0, acc0, false, false);
        }
#pragma unroll
        for (int t = 24; t < 48; ++t) {
            v2f b = *(const v2f*)(brow + 4 * t);
            acc1 = __builtin_amdgcn_wmma_f32_16x16x4_f32(false, areg[t], false, b,
                                                         (short)0, acc1, false, false);
        }

        const float cn = c2arr[nlane];
#pragma unroll
        for (int r = 0; r < 8; ++r) {
            float s = fmaf(-2.f, acc0[r] + acc1[r], cn);  // ||c||^2 - 2 x.c  (x^2 const per row)
            if (s < best[r]) { best[r] = s; bidx[r] = nlane; }  // ascending n => argmin tie-break
        }
        __syncthreads();   // all waves done reading blds[it&1] before it is overwritten
    }

    // Dump per-lane winners; lanes 0-15 of slot r = row rt*16+r, lanes 16-31 = row rt*16+r+8.
#pragma unroll
    for (int r = 0; r < 8; ++r) {
        redS[(wv * 8 + r) * 32 + lane] = best[r];
        redI[(wv * 8 + r) * 32 + lane] = bidx[r];
    }
    __syncthreads();

    if (tid < M_TILE) {
        int row0 = tid;
        int rt2 = row0 >> 4;
        int m2  = row0 & 15;
        int r   = m2 & 7;
        int lo  = (m2 >> 3) * 16;
        float bs = 3.4e38f;
        int   bi = 0x7fffffff;
        for (int c4 = 0; c4 < 4; ++c4) {
            int base = ((rt2 * 4 + c4) * 8 + r) * 32 + lo;
            for (int ln = 0; ln < 16; ++ln) {
                float s = redS[base + ln];
                int   i = redI[base + ln];
                if (s < bs || (s == bs && i < bi)) { bs = s; bi = i; }
            }
        }
        rowIdx[row0] = bi;
    }
    __syncthreads();

    // Output: unblockify(codebook[idx]) uses SINGLE-blockify indexing: l=(hb,wb), d=(p,c).
    for (int e = tid; e < M_TILE * D_FEAT; e += THREADS) {
        int lr = e / D_FEAT;
        int d  = e - lr * D_FEAT;
        int l  = lbase + lr;
        int p  = d / 3;
        int c  = d - p * 3;
        int hb = l >> 8;
        int wb = l & 255;
        out[((hb * 8 + (p >> 3)) * W_IMG + (wb * 8 + (p & 7))) * C_IMG + c] =
            cb[rowIdx[lr] * D_FEAT + d];
    }
}

extern "C" void kernel_launch(void* const* d_in, const int* in_sizes, int n_in,
                              void* d_out, int out_size, void* d_ws, size_t ws_size,
                              hipStream_t stream) {
    const float* image = (const float*)d_in[0];   // [2048,2048,3] f32
    const float* cb    = (const float*)d_in[1];   // [4096,64,3]   f32
    float* out = (float*)d_out;
    float* c2  = (float*)d_ws;                    // 4096 floats scratch

    vq_c2_kernel<<<N_CODE / 256, 256, 0, stream>>>(cb, c2);
    vq_main_kernel<<<L_ROWS / M_TILE, THREADS, 0, stream>>>(image, cb, c2, out);
}